// DecoderCell_90666759618584
// MI455X (gfx1250) — compile-verified
//
#include <hip/hip_runtime.h>
#include <math.h>

// ---- problem dims (match reference) ----
#define B_   64
#define C_   8
#define N_   512
#define E_   512
#define H_   8
#define DH_  64
#define EP2  514          // E + 2
#define NEGV (-1e9f)
#define CLIPV 10.0f
#define INV_SQRT_E 0.04419417382415922f   // 1/sqrt(512)

typedef __attribute__((ext_vector_type(2))) float v2f;
typedef __attribute__((ext_vector_type(8))) float v8f;
typedef int vsi4 __attribute__((vector_size(4 * sizeof(int))));

// gfx1250 async global->LDS copies (ASYNCcnt-tracked DMA, bypasses VGPRs)
// Builtin param types per hipcc diagnostic: b32 wants int*, b128 wants vsi4*.
// Arg order mirrors __builtin_amdgcn_global_load_lds: (global src, lds dst, off, cpol).
__device__ __forceinline__ void async_b32(const float* g, float* l) {
    __builtin_amdgcn_global_load_async_to_lds_b32((int*)g, (int*)l, 0, 0);
}
__device__ __forceinline__ void async_b128(const float* g, float* l) {
    __builtin_amdgcn_global_load_async_to_lds_b128((vsi4*)g, (vsi4*)l, 0, 0);
}

// ------------------------------------------------------------------
// 1) wcol[e] = sum_i Wk2[i,e]   (row-sum trick: k2sum = node @ wcol)
// ------------------------------------------------------------------
__global__ void colsum_k2(const float* __restrict__ Wk2, float* __restrict__ wcol) {
    int e = blockIdx.x * blockDim.x + threadIdx.x;
    if (e < E_) {
        float s = 0.f;
        for (int i = 0; i < E_; ++i) s += Wk2[(size_t)i * E_ + e];
        wcol[e] = s;
    }
}

// ------------------------------------------------------------------
// 2) k2sum[b,n] = node[b,n,:] . wcol   (wave-per-row, coalesced)
// ------------------------------------------------------------------
__global__ void k2sum_k(const float* __restrict__ node, const float* __restrict__ wcol,
                        float* __restrict__ k2) {
    int row = blockIdx.x * 8 + (threadIdx.x >> 5);   // row in [0, B*N)
    int l   = threadIdx.x & 31;
    const float* r = node + (size_t)row * E_;
    float s = 0.f;
    for (int e = l; e < E_; e += 32) s += r[e] * wcol[e];
    for (int off = 16; off > 0; off >>= 1) s += __shfl_down(s, off, 32);
    if (l == 0) k2[row] = s;
}

// ------------------------------------------------------------------
// 3) Q1[b,c,i] = (g[b] @ Wqf.T)[i] + (sc[b,c] @ Wqs.T)[i]
//    u[b,c]    = sum_i tanh(Q1) * v[i]
// ------------------------------------------------------------------
__global__ void q1_k(const float* __restrict__ g, const float* __restrict__ sc,
                     const float* __restrict__ Wqf, const float* __restrict__ Wqs,
                     const float* __restrict__ vvec,
                     float* __restrict__ Q1, float* __restrict__ u) {
    int bc = blockIdx.x;           // b*C + c
    int b  = bc / C_;
    __shared__ float red[256];
    const float* grow = g  + (size_t)b  * E_;
    const float* srow = sc + (size_t)bc * EP2;
    float part = 0.f;
    for (int i = threadIdx.x; i < E_; i += 256) {
        const float* wf = Wqf + (size_t)i * E_;
        const float* ws = Wqs + (size_t)i * EP2;
        float acc = 0.f;
        for (int e = 0; e < E_;  ++e) acc += grow[e] * wf[e];
        for (int j = 0; j < EP2; ++j) acc += srow[j] * ws[j];
        Q1[(size_t)bc * E_ + i] = acc;
        part += tanhf(acc) * vvec[i];
    }
    red[threadIdx.x] = part;
    __syncthreads();
    for (int s = 128; s > 0; s >>= 1) {
        if (threadIdx.x < s) red[threadIdx.x] += red[threadIdx.x + s];
        __syncthreads();
    }
    if (threadIdx.x == 0) u[bc] = red[0];
}

// ------------------------------------------------------------------
// 4) qt[b, r=(c*H+h), e] = (1/sqrt(dh)) * sum_d Q1[b,c,h*64+d] * Wk1[h*64+d, e]
// ------------------------------------------------------------------
__global__ void qtilde_k(const float* __restrict__ Q1, const float* __restrict__ Wk1,
                         float* __restrict__ qt) {
    int br = blockIdx.x;           // b*64 + r
    int b = br >> 6, r = br & 63;
    int c = r >> 3,  h = r & 7;
    __shared__ float qrow[DH_];
    if (threadIdx.x < DH_)
        qrow[threadIdx.x] = Q1[((size_t)(b * C_ + c)) * E_ + h * DH_ + threadIdx.x];
    __syncthreads();
    for (int e = threadIdx.x; e < E_; e += 256) {
        float acc = 0.f;
        for (int d = 0; d < DH_; ++d)
            acc += qrow[d] * Wk1[(size_t)(h * DH_ + d) * E_ + e];
        qt[(size_t)br * E_ + e] = acc * 0.125f;   // 1/sqrt(64)
    }
}

// ------------------------------------------------------------------
// 5/7) Per-batch GEMM with f32 WMMA + async double-buffered LDS staging.
//   Out[b][64 x 512] = A[b][64 x 512] * Bel,  Bel[k][n] = TRANSB ? Bm[b][n][k]
//                                                                : Bm[b][k][n]
//   Block = 256 threads (8 waves). Block covers a 256-wide N chunk; each wave
//   owns a 32-col strip (2 n-tiles) x 4 m-tiles  -> 8 WMMA per 6 LDS pair-loads.
// ------------------------------------------------------------------
#define ASTRIDE 65    // [k][m] padded
#define BSTRIDE 260   // [k][n] padded; 260*4 bytes = multiple of 16 (b128-safe)

template <int TRANSB>
__global__ void gemm64_wmma(const float* __restrict__ A, const float* __restrict__ Bm,
                            float* __restrict__ Out) {
    __shared__ __align__(16) float As[2][32 * ASTRIDE];
    __shared__ __align__(16) float Bs[2][32 * BSTRIDE];

    int b    = blockIdx.x >> 1;
    int n0   = (blockIdx.x & 1) * 256;
    int tid  = threadIdx.x;
    int lane = tid & 31;
    int w    = tid >> 5;                 // wave id 0..7
    int koff = (lane >= 16) ? 2 : 0;     // ISA 16x4 f32 A layout: K pairs {0,1}|{2,3}
    int lm   = lane & 15;

    const float* Ab = A  + (size_t)b * 64 * E_;
    const float* Bb = Bm + (size_t)b * N_ * E_;

    // async issues per thread (== per wave instruction count) for one stage
    constexpr int PEND = 8 + (TRANSB ? 32 : 8);

    v8f acc[4][2];
    #pragma unroll
    for (int mt = 0; mt < 4; ++mt)
        #pragma unroll
        for (int nt = 0; nt < 2; ++nt)
            acc[mt][nt] = (v8f){0.f, 0.f, 0.f, 0.f, 0.f, 0.f, 0.f, 0.f};

    // ---- stage one 32-deep K chunk into buffer `buf` ----
    auto stage = [&](int buf, int k0) {
        {   // A: 64 m x 32 k, transposed -> As[k][m]
            int kk = tid & 31, mm = tid >> 5;         // mm 0..7
            #pragma unroll
            for (int p = 0; p < 8; ++p)
                async_b32(Ab + (size_t)(mm + p * 8) * E_ + k0 + kk,
                          &As[buf][kk * ASTRIDE + mm + p * 8]);
        }
        if (TRANSB) {   // Bel[k][n] = Bb[n][k] (transpose via per-element b32)
            int kk = tid & 31, nn = tid >> 5;         // nn 0..7
            #pragma unroll
            for (int p = 0; p < 32; ++p)
                async_b32(Bb + (size_t)(n0 + nn + p * 8) * E_ + k0 + kk,
                          &Bs[buf][kk * BSTRIDE + nn + p * 8]);
        } else {        // Bel[k][n] = Bb[k][n] (row-contiguous -> b128)
            int e4 = (tid & 63) * 4, kk = tid >> 6;   // kk 0..3
            #pragma unroll
            for (int p = 0; p < 8; ++p)
                async_b128(Bb + (size_t)(k0 + kk + p * 4) * E_ + n0 + e4,
                           &Bs[buf][(kk + p * 4) * BSTRIDE + e4]);
        }
    };

    stage(0, 0);
    for (int c = 0; c < 16; ++c) {
        int cur = c & 1;
        if (c < 15) {
            stage(cur ^ 1, (c + 1) * 32);
            __builtin_amdgcn_s_wait_asynccnt(PEND);   // drain current buffer only
        } else {
            __builtin_amdgcn_s_wait_asynccnt(0);
        }
        __syncthreads();                              // all waves: buffer `cur` ready

        #pragma unroll
        for (int ks = 0; ks < 32; ks += 4) {
            v2f bf0, bf1;
            bf0.x = Bs[cur][(ks + koff)     * BSTRIDE + w * 32 + lm];
            bf0.y = Bs[cur][(ks + koff + 1) * BSTRIDE + w * 32 + lm];
            bf1.x = Bs[cur][(ks + koff)     * BSTRIDE + w * 32 + 16 + lm];
            bf1.y = Bs[cur][(ks + koff + 1) * BSTRIDE + w * 32 + 16 + lm];
            #pragma unroll
            for (int mt = 0; mt < 4; ++mt) {
                v2f af;
                af.x = As[cur][(ks + koff)     * ASTRIDE + mt * 16 + lm];
                af.y = As[cur][(ks + koff + 1) * ASTRIDE + mt * 16 + lm];
                acc[mt][0] = __builtin_amdgcn_wmma_f32_16x16x4_f32(
                    false, af, false, bf0, (short)0, acc[mt][0], false, false);
                acc[mt][1] = __builtin_amdgcn_wmma_f32_16x16x4_f32(
                    false, af, false, bf1, (short)0, acc[mt][1], false, false);
            }
        }
        __syncthreads();   // done reading `cur` before it is re-staged
    }

    // C/D layout: VGPR g, lanes 0-15 -> row g, lanes 16-31 -> row g+8; col = lane&15
    int mhalf = (lane >= 16) ? 8 : 0;
    #pragma unroll
    for (int mt = 0; mt < 4; ++mt)
        #pragma unroll
        for (int nt = 0; nt < 2; ++nt)
            #pragma unroll
            for (int gi = 0; gi < 8; ++gi)
                Out[(size_t)b * 64 * E_ + (size_t)(mt * 16 + gi + mhalf) * E_ +
                    n0 + w * 32 + nt * 16 + lm] = acc[mt][nt][gi];
}

// ------------------------------------------------------------------
// 6) masked row softmax over N, in place (S -> attn)
// ------------------------------------------------------------------
__global__ void softmax_k(float* __restrict__ S, const unsigned char* __restrict__ mask) {
    int row = blockIdx.x;              // b*64 + (c*8 + h)
    int b = row >> 6, r = row & 63, c = r >> 3;
    const unsigned char* m = mask + ((size_t)(b * C_ + c)) * N_;
    float* s = S + (size_t)row * N_;
    int t = threadIdx.x;
    __shared__ float red[256];

    float x0 = m[t]       ? NEGV : s[t];
    float x1 = m[t + 256] ? NEGV : s[t + 256];

    red[t] = fmaxf(x0, x1);
    __syncthreads();
    for (int st = 128; st > 0; st >>= 1) {
        if (t < st) red[t] = fmaxf(red[t], red[t + st]);
        __syncthreads();
    }
    float mx = red[0];
    __syncthreads();

    float e0 = expf(x0 - mx), e1 = expf(x1 - mx);
    red[t] = e0 + e1;
    __syncthreads();
    for (int st = 128; st > 0; st >>= 1) {
        if (t < st) red[t] += red[t + st];
        __syncthreads();
    }
    float inv = 1.f / red[0];
    s[t]       = e0 * inv;
    s[t + 256] = e1 * inv;
}

// ------------------------------------------------------------------
// 8) pointer logits -> d_out[0 .. B*C*N)
// ------------------------------------------------------------------
__global__ void logits_k(const float* __restrict__ u, const float* __restrict__ k2,
                         const unsigned char* __restrict__ mask, float* __restrict__ out) {
    int idx = blockIdx.x * 256 + threadIdx.x;    // [0, B*C*N)
    int bc = idx / N_, n = idx % N_, b = bc / C_;
    float val;
    if (mask[idx]) val = NEGV;
    else           val = CLIPV * tanhf(u[bc] * k2[(size_t)b * N_ + n] * INV_SQRT_E);
    out[idx] = val;
}

// ------------------------------------------------------------------
// 9) heads[b,c,i] = sum_e t[b, c*8 + i/64, e] * Wv[i,e]
// ------------------------------------------------------------------
__global__ void heads_k(const float* __restrict__ tmat, const float* __restrict__ Wv,
                        float* __restrict__ heads) {
    int bc = blockIdx.x;
    int b = bc / C_, c = bc % C_;
    for (int i = threadIdx.x; i < E_; i += 256) {
        int h = i >> 6;
        const float* trow = tmat + ((size_t)(b * 64 + c * H_ + h)) * E_;
        const float* wrow = Wv + (size_t)i * E_;
        float acc = 0.f;
        for (int e = 0; e < E_; ++e) acc += trow[e] * wrow[e];
        heads[(size_t)bc * E_ + i] = acc;
    }
}

// ------------------------------------------------------------------
// 10) Q3[b,c,i] = sum_j heads[b,c,j] * Wout[i,j] -> d_out[B*C*N ..)
// ------------------------------------------------------------------
__global__ void q3_k(const float* __restrict__ heads, const float* __restrict__ Wout,
                     float* __restrict__ out) {
    int bc = blockIdx.x;
    const float* hrow = heads + (size_t)bc * E_;
    for (int i = threadIdx.x; i < E_; i += 256) {
        const float* wrow = Wout + (size_t)i * E_;
        float acc = 0.f;
        for (int j = 0; j < E_; ++j) acc += hrow[j] * wrow[j];
        out[(size_t)bc * E_ + i] = acc;
    }
}

extern "C" void kernel_launch(void* const* d_in, const int* in_sizes, int n_in,
                              void* d_out, int out_size, void* d_ws, size_t ws_size,
                              hipStream_t stream) {
    const float* node          = (const float*)d_in[0];
    const float* g             = (const float*)d_in[1];
    const float* sc            = (const float*)d_in[2];
    const unsigned char* mask  = (const unsigned char*)d_in[3];
    const float* Wk1           = (const float*)d_in[4];
    const float* Wv            = (const float*)d_in[5];
    const float* Wk2           = (const float*)d_in[6];
    const float* Wqf           = (const float*)d_in[7];
    const float* Wout          = (const float*)d_in[8];
    const float* Wqs           = (const float*)d_in[9];
    const float* vvec          = (const float*)d_in[10];

    float* ws   = (float*)d_ws;
    float* Q1   = ws;                    // B*C*E           = 262144
    float* u    = Q1   + 262144;         // B*C             = 512
    float* wcol = u    + 512;            // E               = 512
    float* k2   = wcol + 512;            // B*N             = 32768
    float* qt   = k2   + 32768;          // B*64*E          = 2097152
    float* S    = qt   + 2097152;        // B*64*N          = 2097152
    float* tmat = S    + 2097152;        // B*64*E          = 2097152
    float* hd   = tmat + 2097152;        // B*C*E           = 262144

    float* logits = (float*)d_out;       // B*C*N = 262144
    float* Q3     = logits + B_ * C_ * N_;

    hipLaunchKernelGGL(colsum_k2,   dim3(2),            dim3(256), 0, stream, Wk2, wcol);
    hipLaunchKernelGGL(k2sum_k,     dim3(B_ * N_ / 8),  dim3(256), 0, stream, node, wcol, k2);
    hipLaunchKernelGGL(q1_k,        dim3(B_ * C_),      dim3(256), 0, stream, g, sc, Wqf, Wqs, vvec, Q1, u);
    hipLaunchKernelGGL(qtilde_k,    dim3(B_ * 64),      dim3(256), 0, stream, Q1, Wk1, qt);
    gemm64_wmma<1><<<dim3(B_ * 2), dim3(256), 0, stream>>>(qt, node, S);
    hipLaunchKernelGGL(softmax_k,   dim3(B_ * 64),      dim3(256), 0, stream, S, mask);
    hipLaunchKernelGGL(logits_k,    dim3(B_ * C_ * N_ / 256), dim3(256), 0, stream, u, k2, mask, logits);
    gemm64_wmma<0><<<dim3(B_ * 2), dim3(256), 0, stream>>>(S, node, tmat);
    hipLaunchKernelGGL(heads_k,     dim3(B_ * C_),      dim3(256), 0, stream, tmat, Wv, hd);
    hipLaunchKernelGGL(q3_k,        dim3(B_ * C_),      dim3(256), 0, stream, hd, Wout, Q3);
}